// TransMorph_DS_MoSE_84593675862315
// MI455X (gfx1250) — compile-verified
//
#include <hip/hip_runtime.h>
#include <math.h>

// Problem geometry (TransMorph MoSE head)
#define N_EXPERTS 48
#define N_CAT     30
#define N_VOX     13440                 // 24*28*20
#define N_ROW     (N_CAT * N_VOX)       // 403200 rows of 48 experts
#define BLK       256

// ---------------------------------------------------------------------------
// CDNA5 async global->LDS path (gfx1250).
// Signature (from clang diagnostic): first param is v4i addrspace(1)* (global
// source), second is the LDS destination; then imm offset + imm cpol.
// ---------------------------------------------------------------------------
#if defined(__HIP_DEVICE_COMPILE__) && __has_builtin(__builtin_amdgcn_global_load_async_to_lds_b128)
#define USE_ASYNC 1
typedef int v4i_t __attribute__((ext_vector_type(4)));
typedef __attribute__((address_space(1))) v4i_t Gv4i;   // global b128
typedef __attribute__((address_space(3))) v4i_t Lv4i;   // LDS b128
#else
#define USE_ASYNC 0
#endif

__device__ __forceinline__ void wait_async0() {
#if defined(__HIP_DEVICE_COMPILE__)
#if __has_builtin(__builtin_amdgcn_s_wait_asynccnt)
  __builtin_amdgcn_s_wait_asynccnt(0);
#else
  asm volatile("s_wait_asynccnt 0" ::: "memory");
#endif
#endif
}

// ---------------------------------------------------------------------------
// Kernel 1: per-row top-K masked dot product.
//   row = c*13440 + voxel  in [0, 403200)
//   element (k,row) lives at flat index k*N_ROW + row  -> wave32-coalesced.
// shape_experts tile (48 x 256 rows) is staged once into LDS (async b128) and
// reused by both the m-pass and the f-pass (saves a full 77 MB re-read).
// Top-K selection: branch-free pairwise ranking, |abs| folded into the
// v_cmp source modifiers; ties resolved toward the lower index exactly as
// jax.lax.top_k does.
// ---------------------------------------------------------------------------
__global__ __launch_bounds__(BLK) void moe_map_kernel(
    const float* __restrict__ gm, const float* __restrict__ gf,
    const float* __restrict__ se, const int* __restrict__ kptr,
    float* __restrict__ out /* [2 * N_ROW]: m map then f map */) {
  __shared__ float s_tile[N_EXPERTS * BLK];  // 48 KB
  const int tid  = threadIdx.x;
  const int row0 = blockIdx.x * BLK;
  const int row  = row0 + tid;
  const int K    = *kptr;

  // Early cache hints for the gating streams (emits global_prefetch_b8).
  __builtin_prefetch(gm + row, 0, 0);
  __builtin_prefetch(gf + row, 0, 0);

#if USE_ASYNC
  // 48*256 floats = 3072 x b128 transfers = 12 rounds of 256 lanes.
  #pragma unroll
  for (int it = 0; it < (N_EXPERTS * BLK) / (BLK * 4); ++it) {
    const int e4 = (it * BLK + tid) * 4;     // float index into tile, %4==0
    const int k  = e4 >> 8;                  // 256 floats per expert slice
    const int r  = e4 & (BLK - 1);
    const float* gp = se + (size_t)k * N_ROW + row0 + r;   // 16B aligned
    __builtin_amdgcn_global_load_async_to_lds_b128((Gv4i*)gp,
                                                   (Lv4i*)&s_tile[e4], 0, 0);
  }
  wait_async0();
#else
  #pragma unroll
  for (int k = 0; k < N_EXPERTS; ++k)
    s_tile[k * BLK + tid] = se[(size_t)k * N_ROW + row0 + tid];
#endif
  __syncthreads();

  const float* gbase[2] = {gm, gf};
  #pragma unroll 1
  for (int pass = 0; pass < 2; ++pass) {
    const float* __restrict__ g = gbase[pass];

    // 48 coalesced, single-use (nontemporal) gating loads into registers.
    float gv[N_EXPERTS];
    #pragma unroll
    for (int k = 0; k < N_EXPERTS; ++k)
      gv[k] = __builtin_nontemporal_load(g + (size_t)k * N_ROW + row);

    // Pairwise rank: rk[k] = #elements strictly above k in top_k order.
    int rk[N_EXPERTS];
    #pragma unroll
    for (int k = 0; k < N_EXPERTS; ++k) rk[k] = 0;
    #pragma unroll
    for (int j = 0; j < N_EXPERTS - 1; ++j) {
      #pragma unroll
      for (int k = j + 1; k < N_EXPERTS; ++k) {
        // j<k: on ties the lower index (j) wins, matching jax.lax.top_k.
        const int jw = (fabsf(gv[j]) >= fabsf(gv[k])) ? 1 : 0;
        rk[k] += jw;
        rk[j] += 1 - jw;
      }
    }

    // Masked dot with the LDS-staged shape_experts tile (bank-conflict-free).
    float acc = 0.0f;
    #pragma unroll
    for (int k = 0; k < N_EXPERTS; ++k)
      acc = (rk[k] < K) ? fmaf(gv[k], s_tile[k * BLK + tid], acc) : acc;

    out[(size_t)pass * N_ROW + row] = acc;
  }
}

// ---------------------------------------------------------------------------
// Kernel 2: in-place softmax over the 30 categories of each spatial voxel,
// for both maps. Coalesced over voxels; v_exp_f32 via __expf.
// ---------------------------------------------------------------------------
__global__ __launch_bounds__(BLK) void softmax_kernel(float* __restrict__ out) {
  const int v   = blockIdx.x * BLK + threadIdx.x;
  const int map = blockIdx.y;
  if (v >= N_VOX) return;
  float* base = out + (size_t)map * N_ROW + v;

  float vals[N_CAT];
  float mx = -3.402823466e38f;
  #pragma unroll
  for (int c = 0; c < N_CAT; ++c) {
    vals[c] = base[(size_t)c * N_VOX];
    mx = fmaxf(mx, vals[c]);
  }
  float s = 0.0f;
  #pragma unroll
  for (int c = 0; c < N_CAT; ++c) {
    vals[c] = __expf(vals[c] - mx);
    s += vals[c];
  }
  const float inv = 1.0f / s;
  #pragma unroll
  for (int c = 0; c < N_CAT; ++c)
    base[(size_t)c * N_VOX] = vals[c] * inv;
}

// ---------------------------------------------------------------------------
extern "C" void kernel_launch(void* const* d_in, const int* in_sizes, int n_in,
                              void* d_out, int out_size, void* d_ws, size_t ws_size,
                              hipStream_t stream) {
  (void)in_sizes; (void)n_in; (void)out_size; (void)d_ws; (void)ws_size;
  const float* gm = (const float*)d_in[0];   // gating_m   (48,30,24,28,20) f32
  const float* gf = (const float*)d_in[1];   // gating_f
  const float* se = (const float*)d_in[2];   // shape_experts
  const int*   kp = (const int*)d_in[3];     // k_experts (scalar on device)
  float* out = (float*)d_out;                // (2,1,30,24,28,20) f32

  moe_map_kernel<<<dim3(N_ROW / BLK), dim3(BLK), 0, stream>>>(gm, gf, se, kp, out);
  softmax_kernel<<<dim3((N_VOX + BLK - 1) / BLK, 2), dim3(BLK), 0, stream>>>(out);
}